// torchkb_nufft_75522704933326
// MI455X (gfx1250) — compile-verified
//
#include <hip/hip_runtime.h>
#include <hip/hip_bf16.h>

// ---------------------------------------------------------------------------
// NUFFT forward (torchkbnufft-style) for MI455X / gfx1250.
//   img : (1,8,256,256) complex64   -> d_in[0] as interleaved float pairs
//   trj : (1,48,1024,2) float32     -> d_in[1]
//   out : (1,8,48,1024) complex64   -> d_out as interleaved float pairs
//
// Pipeline:
//   (1) build DFT tables E[k,x] = (-1)^{k+x} * exp(-2*pi*i*k*x/512)
//       (signs implement ifftshift/fftshift; ortho 1/512 folded into stage-1)
//   (2) deapodize image into split real/imag planes
//   (3) grid = E * A * E^T via fp32 WMMA (16x16x4), complex as 4 real matmuls
//   (4) Kaiser-Bessel 6x6 interpolation gather per trajectory point
// ---------------------------------------------------------------------------

typedef __attribute__((ext_vector_type(2))) float v2f;
typedef __attribute__((ext_vector_type(8))) float v8f;

#define NUFFT_PI   3.14159265358979323846f
#define NUFFT_J    6
#define NUFFT_ALPHA (2.34f * 6.0f)           // 14.04
#define GRID_N     512
#define IMG_N      256
#define NCH        8
#define NPTS       (48 * 1024)               // 49152

// ---------------- workspace layout (in floats) ------------------------------
// CE  [512*256]  stage-1 A table, real   (scaled by 1/512)
// SE  [512*256]  stage-1 A table, imag   (scaled by 1/512)
// CET [256*512]  stage-2 B table, real
// SET [256*512]  stage-2 B table, imag
// Ar  [8*256*256] deapodized image real
// Ai  [8*256*256] deapodized image imag
// Tr  [8*512*256] intermediate real
// Ti  [8*512*256] intermediate imag
// G   [8*512*512] interleaved complex grid (float2)
#define OFF_CE  0
#define OFF_SE  (OFF_CE  + 512*256)
#define OFF_CET (OFF_SE  + 512*256)
#define OFF_SET (OFF_CET + 256*512)
#define OFF_AR  (OFF_SET + 256*512)
#define OFF_AI  (OFF_AR  + NCH*256*256)
#define OFF_TR  (OFF_AI  + NCH*256*256)
#define OFF_TI  (OFF_TR  + NCH*512*256)
#define OFF_G   (OFF_TI  + NCH*512*256)

// ---------------- device helpers --------------------------------------------

__device__ __forceinline__ float nufft_i0(float z) {
  // Abramowitz & Stegun 9.8.1 / 9.8.2 polynomial approximations.
  if (z < 3.75f) {
    float t = z * (1.0f / 3.75f);
    t *= t;
    return 1.0f + t*(3.5156229f + t*(3.0899424f + t*(1.2067492f +
               t*(0.2659732f + t*(0.0360768f + t*0.0045813f)))));
  }
  float t = 3.75f / z;
  float p = 0.39894228f + t*(0.01328592f + t*(0.00225319f + t*(-0.00157565f +
            t*(0.00916281f + t*(-0.02057706f + t*(0.02635537f +
            t*(-0.01647633f + t*0.00392377f)))))));
  return (__expf(z) * __frsqrt_rn(z)) * p;
}

__device__ __forceinline__ float nufft_kbw(float u) {
  // I0(alpha*sqrt(1 - (2u/J)^2)) for |2u/J| < 1 else 0
  float x = 1.0f - (u * (2.0f / NUFFT_J)) * (u * (2.0f / NUFFT_J));
  if (x <= 0.0f) return 0.0f;
  return nufft_i0(NUFFT_ALPHA * sqrtf(x));
}

__device__ __forceinline__ float nufft_deapod1(int n) {
  // J * sinh(s)/s, s = sqrt(alpha^2 - (pi*J*f)^2), f = (n-128)/512.
  // (pi*J*f)^2 <= 22.2 < alpha^2 = 197.1, so z is always pure-imaginary.
  float f = ((float)n - 128.0f) * (1.0f / 512.0f);
  float t = NUFFT_PI * (float)NUFFT_J * f;
  float s = sqrtf(NUFFT_ALPHA * NUFFT_ALPHA - t * t);
  float sh = 0.5f * (__expf(s) - __expf(-s));
  return (float)NUFFT_J * sh / s;
}

// ---------------- (1) DFT tables --------------------------------------------

__global__ void nufft_tables_k(float* __restrict__ CE, float* __restrict__ SE,
                               float* __restrict__ CET, float* __restrict__ SET) {
  int idx = blockIdx.x * blockDim.x + threadIdx.x;   // 512*256 entries
  if (idx >= 512 * 256) return;
  int k  = idx >> 8;        // frequency row 0..511
  int x  = idx & 255;       // image row 0..255
  int xg = x + 128;         // position in padded 512 grid
  // exact integer phase reduction: (k*xg) mod 512 keeps cosf argument small
  int m = (k * xg) & 511;
  float ang = -(float)m * (2.0f * NUFFT_PI / 512.0f);
  float sgn = ((k + xg) & 1) ? -1.0f : 1.0f;         // ifftshift+fftshift signs
  float cr = sgn * __cosf(ang);
  float ci = sgn * __sinf(ang);
  CE [k * 256 + x] = cr * (1.0f / 512.0f);           // ortho norm folded here
  SE [k * 256 + x] = ci * (1.0f / 512.0f);
  CET[x * 512 + k] = cr;
  SET[x * 512 + k] = ci;
}

// ---------------- (2) deapodization -----------------------------------------

__global__ void nufft_deapod_k(const float* __restrict__ img,
                               float* __restrict__ Ar, float* __restrict__ Ai) {
  int idx = blockIdx.x * blockDim.x + threadIdx.x;   // 8*256*256
  if (idx >= NCH * 256 * 256) return;
  int rem = idx & 65535;
  int x = rem >> 8, y = rem & 255;
  float inv = 1.0f / (nufft_deapod1(x) * nufft_deapod1(y));
  float re = img[2 * idx + 0];
  float im = img[2 * idx + 1];
  Ar[idx] = re * inv;
  Ai[idx] = im * inv;
}

// ---------------- (3a) stage 1: T = E * A   (512x256 = 512x256 * 256x256) ----
// One wave per 16x16 complex output tile; 4 fp32 WMMAs per K-step of 4.

__global__ __launch_bounds__(128)
void nufft_stage1_k(const float* __restrict__ CE, const float* __restrict__ SE,
                    const float* __restrict__ Ar, const float* __restrict__ Ai,
                    float* __restrict__ Tr, float* __restrict__ Ti) {
  const int lane = threadIdx.x & 31;
  const int wave = threadIdx.x >> 5;
  const int tm = blockIdx.x;                 // kx tile: 0..31
  const int tn = blockIdx.y * 4 + wave;      // y  tile: 0..15
  const int c  = blockIdx.z;
  const int m  = lane & 15;                  // A: row M ; B/D: col N
  const int kk = (lane >> 4) << 1;           // 0 or 2 (K sub-offset)

  const float* ar = Ar + c * 256 * 256;
  const float* ai = Ai + c * 256 * 256;

  v8f accR = {};
  v8f accI = {};

  for (int k = 0; k < 256; k += 4) {
    // A fragments from E tables (row-major 512x256): lane=M, regs span K
    const int aoff = (tm * 16 + m) * 256 + k + kk;
    v2f aC = { CE[aoff], CE[aoff + 1] };
    v2f aS = { SE[aoff], SE[aoff + 1] };
    // B fragments from image planes (row-major 256x256): lane=N, regs span K
    const int boff = (k + kk) * 256 + tn * 16 + m;
    v2f bR  = { ar[boff], ar[boff + 256] };
    v2f bI  = { ai[boff], ai[boff + 256] };
    v2f nbI = { -bI.x, -bI.y };              // f32 WMMA has no A/B NEG bits

    // (Cr + i*Ci) = (aC + i*aS) * (bR + i*bI)
    accR = __builtin_amdgcn_wmma_f32_16x16x4_f32(false, aC, false, bR,  (short)0, accR, false, false);
    accR = __builtin_amdgcn_wmma_f32_16x16x4_f32(false, aS, false, nbI, (short)0, accR, false, false);
    accI = __builtin_amdgcn_wmma_f32_16x16x4_f32(false, aC, false, bI,  (short)0, accI, false, false);
    accI = __builtin_amdgcn_wmma_f32_16x16x4_f32(false, aS, false, bR,  (short)0, accI, false, false);
  }

  const int nh = lane >> 4;                  // C/D: lanes 16..31 hold M+8
  #pragma unroll
  for (int r = 0; r < 8; ++r) {
    int row = tm * 16 + r + 8 * nh;
    int col = tn * 16 + m;
    Tr[(c * 512 + row) * 256 + col] = accR[r];
    Ti[(c * 512 + row) * 256 + col] = accI[r];
  }
}

// ---------------- (3b) stage 2: G = T * E^T (512x512 = 512x256 * 256x512) ----

__global__ __launch_bounds__(128)
void nufft_stage2_k(const float* __restrict__ Tr, const float* __restrict__ Ti,
                    const float* __restrict__ CET, const float* __restrict__ SET,
                    float2* __restrict__ G) {
  const int lane = threadIdx.x & 31;
  const int wave = threadIdx.x >> 5;
  const int tm = blockIdx.x;                 // kx tile: 0..31
  const int tn = blockIdx.y * 4 + wave;      // ky tile: 0..31
  const int c  = blockIdx.z;
  const int m  = lane & 15;
  const int kk = (lane >> 4) << 1;

  const float* tr = Tr + c * 512 * 256;
  const float* ti = Ti + c * 512 * 256;

  v8f accR = {};
  v8f accI = {};

  for (int k = 0; k < 256; k += 4) {
    const int aoff = (tm * 16 + m) * 256 + k + kk;
    v2f aR = { tr[aoff], tr[aoff + 1] };
    v2f aI = { ti[aoff], ti[aoff + 1] };
    const int boff = (k + kk) * 512 + tn * 16 + m;
    v2f bC  = { CET[boff], CET[boff + 512] };
    v2f bS  = { SET[boff], SET[boff + 512] };
    v2f nbS = { -bS.x, -bS.y };

    accR = __builtin_amdgcn_wmma_f32_16x16x4_f32(false, aR, false, bC,  (short)0, accR, false, false);
    accR = __builtin_amdgcn_wmma_f32_16x16x4_f32(false, aI, false, nbS, (short)0, accR, false, false);
    accI = __builtin_amdgcn_wmma_f32_16x16x4_f32(false, aR, false, bS,  (short)0, accI, false, false);
    accI = __builtin_amdgcn_wmma_f32_16x16x4_f32(false, aI, false, bC,  (short)0, accI, false, false);
  }

  float2* g = G + c * 512 * 512;
  const int nh = lane >> 4;
  #pragma unroll
  for (int r = 0; r < 8; ++r) {
    int row = tm * 16 + r + 8 * nh;
    int col = tn * 16 + m;
    g[row * 512 + col] = make_float2(accR[r], accI[r]);
  }
}

// ---------------- (4) Kaiser-Bessel interpolation ----------------------------
// One thread per trajectory point; weights computed once, 36 taps x 8 channels.

__global__ void nufft_interp_k(const float* __restrict__ trj,
                               const float2* __restrict__ G,
                               float2* __restrict__ out) {
  int p = blockIdx.x * blockDim.x + threadIdx.x;
  if (p >= NPTS) return;

  const float sc = (float)GRID_N / (2.0f * NUFFT_PI);
  float gx = trj[2 * p + 0] * sc + 256.0f;
  float gy = trj[2 * p + 1] * sc + 256.0f;
  float bx = floorf(gx), by = floorf(gy);
  float fx = gx - bx,    fy = gy - by;
  int ibx = (int)bx, iby = (int)by;

  float wx[6], wy[6];
  int   ix[6], iy[6];
  #pragma unroll
  for (int j = 0; j < 6; ++j) {
    wx[j] = nufft_kbw(fx - (float)(j - 2));
    wy[j] = nufft_kbw(fy - (float)(j - 2));
    ix[j] = (ibx + j - 2 + 512) & 511;       // mod 512 (idx in [-2, 515])
    iy[j] = (iby + j - 2 + 512) & 511;
  }

  float accr[NCH], acci[NCH];
  #pragma unroll
  for (int c = 0; c < NCH; ++c) { accr[c] = 0.0f; acci[c] = 0.0f; }

  for (int a = 0; a < 6; ++a) {
    for (int b = 0; b < 6; ++b) {
      float w = wx[a] * wy[b];
      const float2* gp = G + ix[a] * 512 + iy[b];
      #pragma unroll
      for (int c = 0; c < NCH; ++c) {
        float2 v = gp[c * (512 * 512)];
        accr[c] += v.x * w;
        acci[c] += v.y * w;
      }
    }
  }

  #pragma unroll
  for (int c = 0; c < NCH; ++c)
    out[c * NPTS + p] = make_float2(accr[c], acci[c]);
}

// ---------------- launch -----------------------------------------------------

extern "C" void kernel_launch(void* const* d_in, const int* in_sizes, int n_in,
                              void* d_out, int out_size, void* d_ws, size_t ws_size,
                              hipStream_t stream) {
  (void)in_sizes; (void)n_in; (void)out_size; (void)ws_size;
  const float* img = (const float*)d_in[0];   // complex64 interleaved
  const float* trj = (const float*)d_in[1];

  float* ws  = (float*)d_ws;                  // needs ~30 MB
  float* CE  = ws + OFF_CE;
  float* SE  = ws + OFF_SE;
  float* CET = ws + OFF_CET;
  float* SET = ws + OFF_SET;
  float* Ar  = ws + OFF_AR;
  float* Ai  = ws + OFF_AI;
  float* Tr  = ws + OFF_TR;
  float* Ti  = ws + OFF_TI;
  float2* G  = (float2*)(ws + OFF_G);

  nufft_tables_k<<<dim3(512), dim3(256), 0, stream>>>(CE, SE, CET, SET);
  nufft_deapod_k<<<dim3(2048), dim3(256), 0, stream>>>(img, Ar, Ai);
  nufft_stage1_k<<<dim3(32, 4, NCH), dim3(128), 0, stream>>>(CE, SE, Ar, Ai, Tr, Ti);
  nufft_stage2_k<<<dim3(32, 8, NCH), dim3(128), 0, stream>>>(Tr, Ti, CET, SET, G);
  nufft_interp_k<<<dim3((NPTS + 255) / 256), dim3(256), 0, stream>>>(trj, G, (float2*)d_out);
}